// BlockRelu_9844065042554
// MI455X (gfx1250) — compile-verified
//
#include <hip/hip_runtime.h>

// Tensor: (B=64, C=128, H=56, W=56) fp32, contiguous NCHW.
// PLANE = 56*56 = 3136 floats = 784 float4.
#define PLANE   3136u
#define PLANE4  784u

// Block-range partition of the fused grid (256 threads/block, no tails):
//  relu  : ch [0,32)   -> 64*32*784/256  = 6272 blocks
//  zero  : ch [48,64)  -> 64*16*784/256  = 3136 blocks (store-only)
//  pool2 : ch [64,96)  -> 64*32*784/256  = 6272 blocks (1 lane = one 2x2 block)
//  pool4 : ch [96,128) -> 64*32*196/256  = 1568 blocks (1 lane = one 4x4 block)
//  copy  : ch [32,48)  -> 64*16*784/256  = 3136 blocks (async DMA via LDS)
#define E_RELU  6272u
#define E_ZERO  (E_RELU + 3136u)   // 9408
#define E_P2    (E_ZERO + 6272u)   // 15680
#define E_P4    (E_P2 + 1568u)     // 17248
#define E_CPY   (E_P4 + 3136u)     // 20384

__device__ __forceinline__ float relu_mask(float s) {
    // (sign(s)+1)*0.5  ->  1 if s>0, 0.5 if s==0, 0 if s<0
    return (s > 0.f) ? 1.f : ((s == 0.f) ? 0.5f : 0.f);
}

__global__ __launch_bounds__(256)
void block_relu_fused(const float* __restrict__ in, float* __restrict__ out) {
    __shared__ __align__(16) char smem[256 * 16];   // staging for async DMA copy
    const unsigned bid = blockIdx.x;
    const unsigned tid = threadIdx.x;

    const float4* __restrict__ in4  = (const float4*)in;
    float4* __restrict__       out4 = (float4*)out;

    if (bid < E_RELU) {
        // ---- channels [0,32): elementwise relu mask ----
        unsigned v     = bid * 256u + tid;          // float4 index in region
        unsigned plane = v / PLANE4;                // 0..2047  (b*32 + c)
        unsigned q     = v - plane * PLANE4;
        unsigned b     = plane >> 5;
        unsigned c     = plane & 31u;
        unsigned off   = (b * 128u + c) * PLANE4 + q;
        float4 x = in4[off];
        float4 o;
        o.x = relu_mask(x.x) * x.x;
        o.y = relu_mask(x.y) * x.y;
        o.z = relu_mask(x.z) * x.z;
        o.w = relu_mask(x.w) * x.w;
        out4[off] = o;
    } else if (bid < E_ZERO) {
        // ---- channels [48,64): output is zero; never read the input ----
        unsigned v     = (bid - E_RELU) * 256u + tid;
        unsigned plane = v / PLANE4;                // 0..1023  (b*16 + c)
        unsigned q     = v - plane * PLANE4;
        unsigned b     = plane >> 4;
        unsigned c     = 48u + (plane & 15u);
        out4[(b * 128u + c) * PLANE4 + q] = make_float4(0.f, 0.f, 0.f, 0.f);
    } else if (bid < E_P2) {
        // ---- channels [64,96): 2x2 avg-pool sign mask ----
        unsigned t     = (bid - E_ZERO) * 256u + tid;
        unsigned plane = t / 784u;                  // 0..2047
        unsigned blk   = t - plane * 784u;          // 0..783 (28x28 blocks)
        unsigned by    = blk / 28u;
        unsigned bx    = blk - by * 28u;
        unsigned b     = plane >> 5;
        unsigned c     = 64u + (plane & 31u);
        unsigned base  = (b * 128u + c) * PLANE;
        unsigned r0    = base + (2u * by) * 56u + 2u * bx;
        unsigned r1    = r0 + 56u;
        float2 a = *(const float2*)(in + r0);
        float2 d = *(const float2*)(in + r1);
        float  s = (a.x + a.y) + (d.x + d.y);       // sign(sum) == sign(mean)
        float rm = relu_mask(s);
        *(float2*)(out + r0) = make_float2(rm * a.x, rm * a.y);
        *(float2*)(out + r1) = make_float2(rm * d.x, rm * d.y);
    } else if (bid < E_P4) {
        // ---- channels [96,128): 4x4 avg-pool sign mask ----
        unsigned t     = (bid - E_P2) * 256u + tid;
        unsigned plane = t / 196u;                  // 0..2047
        unsigned blk   = t - plane * 196u;          // 0..195 (14x14 blocks)
        unsigned by    = blk / 14u;
        unsigned bx    = blk - by * 14u;
        unsigned b     = plane >> 5;
        unsigned c     = 96u + (plane & 31u);
        unsigned base  = (b * 128u + c) * PLANE;
        unsigned r     = base + (4u * by) * 56u + 4u * bx;
        float4 v0 = *(const float4*)(in + r);
        float4 v1 = *(const float4*)(in + r + 56u);
        float4 v2 = *(const float4*)(in + r + 112u);
        float4 v3 = *(const float4*)(in + r + 168u);
        float s = ((v0.x + v0.y) + (v0.z + v0.w))
                + ((v1.x + v1.y) + (v1.z + v1.w))
                + ((v2.x + v2.y) + (v2.z + v2.w))
                + ((v3.x + v3.y) + (v3.z + v3.w));
        float rm = relu_mask(s);
        *(float4*)(out + r)        = make_float4(rm*v0.x, rm*v0.y, rm*v0.z, rm*v0.w);
        *(float4*)(out + r + 56u)  = make_float4(rm*v1.x, rm*v1.y, rm*v1.z, rm*v1.w);
        *(float4*)(out + r + 112u) = make_float4(rm*v2.x, rm*v2.y, rm*v2.z, rm*v2.w);
        *(float4*)(out + r + 168u) = make_float4(rm*v3.x, rm*v3.y, rm*v3.z, rm*v3.w);
    } else {
        // ---- channels [32,48): identity copy via CDNA5 async DMA path ----
        // global -> LDS -> global, 16B per lane, never touches VGPRs for data.
        unsigned v     = (bid - E_P4) * 256u + tid; // float4 index in region
        unsigned plane = v / PLANE4;                // 0..1023  (b*16 + c)
        unsigned q     = v - plane * PLANE4;
        unsigned b     = plane >> 4;
        unsigned c     = 32u + (plane & 15u);
        unsigned off   = (b * 128u + c) * PLANE4 + q;

        const float4* gsrc = in4 + off;
        float4*       gdst = out4 + off;
        unsigned lds_off = tid * 16u;               // smem is the sole LDS object -> base 0

        // Keep the LDS allocation alive (smem is only touched by the async HW path).
        asm volatile("" :: "v"(smem) : "memory");

        asm volatile("global_load_async_to_lds_b128 %0, %1, off"
                     :: "v"(lds_off), "v"(gsrc) : "memory");
        asm volatile("s_wait_asynccnt 0" ::: "memory");
        asm volatile("global_store_async_from_lds_b128 %0, %1, off"
                     :: "v"(gdst), "v"(lds_off) : "memory");
        asm volatile("s_wait_asynccnt 0" ::: "memory");
    }
}

extern "C" void kernel_launch(void* const* d_in, const int* in_sizes, int n_in,
                              void* d_out, int out_size, void* d_ws, size_t ws_size,
                              hipStream_t stream) {
    (void)in_sizes; (void)n_in; (void)d_ws; (void)ws_size; (void)out_size;
    const float* act = (const float*)d_in[0];
    float*       out = (float*)d_out;
    block_relu_fused<<<dim3(E_CPY), dim3(256), 0, stream>>>(act, out);
}